// LSTCWA_1494648619528
// MI455X (gfx1250) — compile-verified
//
#include <hip/hip_runtime.h>
#include <math.h>

// ---------------------------------------------------------------------------
// LSTCWA, algebraically collapsed:
//   q  = z @ Wq^T                         (WMMA f32 GEMM)
//   qk = q @ Wk ; qt = q @ w2 ; qb2=q.b2  (WMMA f32 GEMMs + tiny dot)
//   r[l,j]   = feats[l*512+j] . qk[l]     (memory-bound matvec pass #1)
//   logits   = (r + h.qt + qb2)/sqrt(D), h = relu(dx*w1_0+dy*w1_1+b1)
//   a[l,j]  += softmax over each window   (overlap -> atomicAdd)
//   u[l]     = sum_j a[l,j]*feats[l*512+j](memory-bound pass #2)
//   out      = (u @ Wv^T) @ Wo^T + bo     (WMMA f32 GEMMs)
// ~0.3 GFLOP + ~130 MB traffic -> HBM-bound (~6us @ 23.3 TB/s), exact fp32.
// ---------------------------------------------------------------------------

constexpr int kD      = 512;    // DIM
constexpr int kL      = 64;     // L_TOK
constexpr int kSeg    = 512;    // patches per segment (N_PATCH / L_TOK)
constexpr int kW      = 64;     // W_SIZE
constexpr int kStride = 32;     // STRIDE
constexpr int kNwin   = 16;     // windows per segment

typedef float v2f __attribute__((ext_vector_type(2)));
typedef float v8f __attribute__((ext_vector_type(8)));

// C[M,N] = X[M,K] * op(W) (+ bias). TRANSB: W is [N,K] row-major (op = W^T);
// else W is [K,N] row-major. One wave32 per 16x16 C tile, K-loop of
// v_wmma_f32_16x16x4_f32. Fragment layouts per CDNA5 ISA 7.12.2:
//   A 16x4 f32: VGPR0 = K{0|2}, VGPR1 = K{1|3}; lanes 0-15 -> M, 16-31 -> M (K+2)
//   C   16x16 : VGPR i -> M = i (lanes 0-15) / i+8 (lanes 16-31), N = lane&15
template<bool TRANSB, bool BIAS>
__global__ __launch_bounds__(32)
void gemm_wmma_f32(const float* __restrict__ X, const float* __restrict__ W,
                   const float* __restrict__ bias, float* __restrict__ C,
                   int M, int N, int K) {
  const int n0   = blockIdx.x * 16;
  const int m0   = blockIdx.y * 16;
  const int t    = threadIdx.x;
  const int half = t >> 4;
  const int r    = t & 15;
  v8f acc = {};
  for (int k0 = 0; k0 < K; k0 += 4) {
    const int ka = k0 + 2 * half;
    v2f a, b;
    a.x = X[(m0 + r) * K + ka];
    a.y = X[(m0 + r) * K + ka + 1];
    if (TRANSB) {
      b.x = W[(n0 + r) * K + ka];
      b.y = W[(n0 + r) * K + ka + 1];
    } else {
      b.x = W[ka * N + (n0 + r)];
      b.y = W[(ka + 1) * N + (n0 + r)];
    }
    acc = __builtin_amdgcn_wmma_f32_16x16x4_f32(false, a, false, b,
                                                (short)0, acc, false, false);
  }
  const int   nc = n0 + r;
  const float bv = BIAS ? bias[nc] : 0.0f;
  const int   mb = m0 + half * 8;
#pragma unroll
  for (int i = 0; i < 8; ++i) {
    C[(mb + i) * N + nc] = acc[i] + bv;
  }
}

// qb2[l] = q[l] . b2   (one wave per l)
__global__ __launch_bounds__(32)
void dotb2_kernel(const float* __restrict__ q, const float* __restrict__ b2,
                  float* __restrict__ qb2) {
  const int l = blockIdx.x;
  const int lane = threadIdx.x;
  float acc = 0.0f;
  for (int p = lane; p < kD; p += 32) acc += q[l * kD + p] * b2[p];
  for (int m = 16; m > 0; m >>= 1) acc += __shfl_xor(acc, m, 32);
  if (lane == 0) qb2[l] = acc;
}

__global__ void zero_kernel(float* __restrict__ p, int n) {
  const int i = blockIdx.x * blockDim.x + threadIdx.x;
  if (i < n) p[i] = 0.0f;
}

// r[l,j] = feats[l*kSeg+j] . qk[l]; 8 rows per 256-thread block, qk[l] in LDS.
__global__ __launch_bounds__(256)
void scores_kernel(const float* __restrict__ feats, const float* __restrict__ qk,
                   float* __restrict__ r) {
  __shared__ float qs[kD];
  const int l     = blockIdx.x >> 6;          // 64 blocks per segment
  const int chunk = (blockIdx.x & 63) * 8;
  for (int i = threadIdx.x; i < kD; i += 256) qs[i] = qk[l * kD + i];
  __syncthreads();
  const int wave = threadIdx.x >> 5;
  const int lane = threadIdx.x & 31;
  const int j    = chunk + wave;
  const float* fr = feats + (size_t)(l * kSeg + j) * kD;
  float acc = 0.0f;
  for (int p = lane; p < kD; p += 32) acc = fmaf(fr[p], qs[p], acc);
  for (int m = 16; m > 0; m >>= 1) acc += __shfl_xor(acc, m, 32);
  if (lane == 0) r[l * kSeg + j] = acc;
}

// One 64-thread block per (l, window): coord mean, positional-MLP dot with qt,
// masked softmax, accumulate attention weight per patch row.
__global__ __launch_bounds__(64)
void window_kernel(const float* __restrict__ coords, const float* __restrict__ r,
                   const float* __restrict__ qt, const float* __restrict__ qb2,
                   const float* __restrict__ w1, const float* __restrict__ b1,
                   float* __restrict__ a) {
  __shared__ float s1[kW];
  __shared__ float s2[kW];
  const int l = blockIdx.x;
  const int n = blockIdx.y;
  const int w = threadIdx.x;
  const int off   = n * kStride + w;
  const bool valid = off < kSeg;
  const int j     = valid ? off : (kSeg - 1);
  const float cx = coords[(size_t)(l * kSeg + j) * 2 + 0];
  const float cy = coords[(size_t)(l * kSeg + j) * 2 + 1];
  const int rem = kSeg - n * kStride;
  const float cnt = (float)(rem < kW ? rem : kW);

  s1[w] = valid ? cx : 0.0f;
  s2[w] = valid ? cy : 0.0f;
  __syncthreads();
  for (int s = 32; s > 0; s >>= 1) {
    if (w < s) { s1[w] += s1[w + s]; s2[w] += s2[w + s]; }
    __syncthreads();
  }
  const float cmx = s1[0] / cnt;
  const float cmy = s2[0] / cnt;
  __syncthreads();

  float logit = -1e9f;
  float ev;
  if (valid) {
    const float dx = cx - cmx, dy = cy - cmy;
    const float* qtl = qt + l * kD;
    float hdot = 0.0f;
    for (int e = 0; e < kD; ++e) {
      float h = fmaf(dx, w1[2 * e], fmaf(dy, w1[2 * e + 1], b1[e]));
      h = fmaxf(h, 0.0f);
      hdot = fmaf(h, qtl[e], hdot);
    }
    logit = (r[l * kSeg + j] + hdot + qb2[l]) * 0.04419417382415922f; // 1/sqrt(512)
  }

  // softmax over the 64 window slots
  s1[w] = logit;
  __syncthreads();
  for (int s = 32; s > 0; s >>= 1) {
    if (w < s) s1[w] = fmaxf(s1[w], s1[w + s]);
    __syncthreads();
  }
  const float mx = s1[0];
  __syncthreads();
  ev = __expf(logit - mx);          // invalid -> exp(-1e9) underflows to 0
  s1[w] = ev;
  __syncthreads();
  for (int s = 32; s > 0; s >>= 1) {
    if (w < s) s1[w] += s1[w + s];
    __syncthreads();
  }
  const float denom = s1[0];
  if (valid) atomicAdd(&a[l * kSeg + j], ev / denom);
}

// u[l,d] = sum_j a[l,j] * feats[l*kSeg+j, d]; a[l] staged in LDS.
__global__ __launch_bounds__(256)
void weighted_sum_kernel(const float* __restrict__ feats,
                         const float* __restrict__ a, float* __restrict__ u) {
  __shared__ float as[kSeg];
  const int l = blockIdx.x;
  const int d = blockIdx.y * 256 + threadIdx.x;
  for (int i = threadIdx.x; i < kSeg; i += 256) as[i] = a[l * kSeg + i];
  __syncthreads();
  float acc = 0.0f;
  const float* fb = feats + (size_t)l * kSeg * kD + d;
  for (int j = 0; j < kSeg; ++j) acc = fmaf(as[j], fb[(size_t)j * kD], acc);
  u[l * kD + d] = acc;
}

extern "C" void kernel_launch(void* const* d_in, const int* in_sizes, int n_in,
                              void* d_out, int out_size, void* d_ws, size_t ws_size,
                              hipStream_t stream) {
  const float* feats  = (const float*)d_in[0];
  const float* coords = (const float*)d_in[1];
  /* d_in[2] = mask: all-False by construction, unused */
  const float* z  = (const float*)d_in[3];
  const float* Wq = (const float*)d_in[4];
  const float* Wk = (const float*)d_in[5];
  const float* Wv = (const float*)d_in[6];
  const float* w1 = (const float*)d_in[7];
  const float* b1 = (const float*)d_in[8];
  const float* w2 = (const float*)d_in[9];
  const float* b2 = (const float*)d_in[10];
  const float* Wo = (const float*)d_in[11];
  const float* bo = (const float*)d_in[12];
  float* out = (float*)d_out;

  // workspace layout (~0.92 MB of fp32)
  float* ws   = (float*)d_ws;
  float* q    = ws;                  // [64,512]
  float* qk   = q   + kL * kD;       // [64,512]
  float* qt   = qk  + kL * kD;       // [64,512]
  float* qb2  = qt  + kL * kD;       // [64]
  float* rsc  = qb2 + kL;            // [64,512] raw scores
  float* aw   = rsc + kL * kSeg;     // [64,512] accumulated attn weights
  float* u    = aw  + kL * kSeg;     // [64,512]
  float* Zbuf = u   + kL * kD;       // [64,512]

  const dim3 gg(kD / 16, kL / 16);   // 16x16 tiles, one wave each

  // q = z @ Wq^T
  gemm_wmma_f32<true, false><<<gg, 32, 0, stream>>>(z, Wq, nullptr, q, kL, kD, kD);
  // qk = q @ Wk ; qt = q @ w2
  gemm_wmma_f32<false, false><<<gg, 32, 0, stream>>>(q, Wk, nullptr, qk, kL, kD, kD);
  gemm_wmma_f32<false, false><<<gg, 32, 0, stream>>>(q, w2, nullptr, qt, kL, kD, kD);
  dotb2_kernel<<<kL, 32, 0, stream>>>(q, b2, qb2);

  zero_kernel<<<(kL * kSeg) / 256, 256, 0, stream>>>(aw, kL * kSeg);
  scores_kernel<<<kL * 64, 256, 0, stream>>>(feats, qk, rsc);
  window_kernel<<<dim3(kL, kNwin), kW, 0, stream>>>(coords, rsc, qt, qb2, w1, b1, aw);
  weighted_sum_kernel<<<dim3(kL, kD / 256), 256, 0, stream>>>(feats, aw, u);

  // Z = u @ Wv^T ; out = Z @ Wo^T + bo
  gemm_wmma_f32<true, false><<<gg, 32, 0, stream>>>(u, Wv, nullptr, Zbuf, kL, kD, kD);
  gemm_wmma_f32<true, true><<<gg, 32, 0, stream>>>(Zbuf, Wo, bo, out, kL, kD, kD);
}